// QuantumInspiredAdapter_83382495084750
// MI455X (gfx1250) — compile-verified
//
#include <hip/hip_runtime.h>

typedef __attribute__((ext_vector_type(2))) float v2f;
typedef __attribute__((ext_vector_type(8))) float v8f;

namespace {
constexpr int kD           = 768;   // input feature dim
constexpr int kNOut        = 16;    // 2 * n_qubits
constexpr int kLdsStride   = kD + 2;  // pad to de-bank-conflict the transposed W
constexpr int kThreads     = 256;   // 8 waves (wave32)
constexpr int kWavesPerBlk = 8;
}

// One wave32 computes a 16-row x 16-col tile of Y = X @ W using
// V_WMMA_F32_16X16X4_F32, then applies tanh/cos/sin epilogue in-register.
__global__ __launch_bounds__(kThreads)
void qadapter_wmma_kernel(const float* __restrict__ x,     // [rows, 768]
                          const float* __restrict__ W,     // [768, 16]
                          const float* __restrict__ bias,  // [16]
                          const float* __restrict__ enc,   // [8, 3]
                          float* __restrict__ out)         // [rows, 16]
{
    // W transposed in LDS: Wt[n][k] = W[k*16 + n]  (48KB + pad, fits 320KB WGP LDS)
    __shared__ float Wt[kNOut * kLdsStride];

    const int tid = threadIdx.x;
    for (int i = tid; i < kD * kNOut; i += kThreads) {
        const int k = i >> 4;
        const int n = i & 15;
        Wt[n * kLdsStride + k] = W[i];
    }
    __syncthreads();

    const int wave = tid >> 5;
    const int lane = tid & 31;
    const int half = lane >> 4;   // 0: lanes 0-15, 1: lanes 16-31
    const int n    = lane & 15;   // matrix row (A) / col (B,C)

    const int  tile = blockIdx.x * kWavesPerBlk + wave;
    const long row0 = (long)tile * 16;

    // A-fragment addressing (32-bit A 16x4): VGPR0 = K+0 (lo half) / K+2 (hi half),
    // VGPR1 = K+1 / K+3  -> contiguous float2 at k + 2*half.
    const float* xrow = x  + (row0 + n) * kD + half * 2;
    // B-fragment mirrors A in K: float2 from transposed W row n.
    const float* wrow = Wt + n * kLdsStride + half * 2;

    v8f acc = {};
#pragma unroll 8
    for (int k = 0; k < kD; k += 4) {
        v2f a = *(const v2f*)(xrow + k);
        v2f b = *(const v2f*)(wrow + k);
        // 8 args: (neg_a, A, neg_b, B, c_mod, C, reuse_a, reuse_b)
        acc = __builtin_amdgcn_wmma_f32_16x16x4_f32(
            false, a, false, b, (short)0, acc, false, false);
    }

    // Epilogue. C/D layout: VGPR r -> row (r + 8*half), col n.
    // col n: qubit q = n>>1; even n -> real = cos(0.5*theta),
    // odd n -> imag = sin(0.5*theta)*cos(phi) with theta from partner lane (n-1).
    const int   q  = n >> 1;
    const float bn = bias[n];
    const float e0 = enc[q * 3 + 0];
    const float e1 = enc[q * 3 + 1];
    const bool  odd = (n & 1) != 0;

#pragma unroll
    for (int r = 0; r < 8; ++r) {
        const float s  = tanhf(acc[r] + bn);
        const float sp = __shfl_xor(s, 1, 32);   // partner column's tanh value
        const float evenVal = __cosf(0.5f * (s + e0));
        const float oddVal  = __sinf(0.5f * (sp + e0)) * __cosf(s + e1);
        const float v = odd ? oddVal : evenVal;
        const long  m = row0 + r + half * 8;
        out[m * kNOut + n] = v;                  // 64B contiguous per half-wave
    }
}

extern "C" void kernel_launch(void* const* d_in, const int* in_sizes, int n_in,
                              void* d_out, int out_size, void* d_ws, size_t ws_size,
                              hipStream_t stream) {
    const float* x   = (const float*)d_in[0];  // [8*4096, 768]
    const float* W   = (const float*)d_in[1];  // [768, 16]
    const float* b   = (const float*)d_in[2];  // [16]
    const float* enc = (const float*)d_in[3];  // [8, 3]
    float* out = (float*)d_out;                // [8*4096, 16]

    const int rows  = in_sizes[0] / kD;        // 32768
    const int tiles = rows / 16;               // 2048
    dim3 grid(tiles / kWavesPerBlk);           // 256 blocks
    dim3 block(kThreads);
    qadapter_wmma_kernel<<<grid, block, 0, stream>>>(x, W, b, enc, out);
}